// HarmfulAttnProcessor_27041114096255
// MI455X (gfx1250) — compile-verified
//
#include <hip/hip_runtime.h>
#include <math.h>
#include <stdint.h>

typedef float v2f __attribute__((ext_vector_type(2)));
typedef float v8f __attribute__((ext_vector_type(8)));
typedef unsigned int v4u __attribute__((ext_vector_type(4)));
typedef int v8i __attribute__((ext_vector_type(8)));
typedef int v4i __attribute__((ext_vector_type(4)));

#define WMMA_F32(a, b, c) \
  __builtin_amdgcn_wmma_f32_16x16x4_f32(false, (a), false, (b), (short)0, (c), false, false)

#if __has_builtin(__builtin_amdgcn_tensor_load_to_lds)
#define HAVE_TDM 1
#else
#define HAVE_TDM 0
#endif

#if __has_builtin(__builtin_amdgcn_s_wait_tensorcnt)
#define WAIT_TENSOR(n) __builtin_amdgcn_s_wait_tensorcnt((unsigned short)(n))
#else
#define WAIT_TENSOR(n) asm volatile("s_wait_tensorcnt %0" ::"i"(n) : "memory")
#endif

static constexpr int BATCH = 16;
static constexpr int SEQ   = 4096;
static constexpr int CH    = 320;   // channels
static constexpr int KTOK  = 77;    // text tokens
static constexpr int DDIM  = 768;   // text embed dim
static constexpr int NH    = 8;
static constexpr int DH    = 40;    // head dim
static constexpr float TAU   = 0.1f;
static constexpr float GAMMA = 1.0f;
static constexpr float NEG_BIG = -3.0e38f;

// ---------------------------------------------------------------------------
// TDM helper: 2-D tile (fp32) global -> LDS, D# per CDNA5 ISA ch.8.
// pad_bits pre-encodes {pad_enable, pad_interval, pad_amount} for group1 word0.
// This toolchain exposes the 6-arg builtin: (v4u g0, v8i g1, v4i g2, v4i g3,
// v8i g23hi, i32 cpol); groups 2/3 are zero (2-D tile, tile_dim2 = 0).
// ---------------------------------------------------------------------------
__device__ __forceinline__ void tdm_load_2d(uint32_t lds_byte, const void* gptr,
                                            uint32_t tile_x, uint32_t tile_y,
                                            uint32_t tensor_x, uint32_t tensor_y,
                                            uint32_t stride_x, uint32_t pad_bits) {
#if HAVE_TDM
  uint64_t ga = (uint64_t)(uintptr_t)gptr;
  v4u g0;
  g0.x = 1u;                                              // count=1 (valid), user mode
  g0.y = lds_byte;                                        // lds_addr
  g0.z = (uint32_t)ga;                                    // global_addr lo
  g0.w = (uint32_t)((ga >> 32) & 0x1FFFFFFu) | (2u << 30);// global_addr hi | type=2
  v8i g1;
  g1[0] = (int)(0x00020000u | pad_bits);                  // wg_mask=0, data_size=4B
  g1[1] = (int)((tensor_x & 0xFFFFu) << 16);              // tensor_dim0[15:0]
  g1[2] = (int)(((tensor_x >> 16) & 0xFFFFu) | ((tensor_y & 0xFFFFu) << 16));
  g1[3] = (int)(((tensor_y >> 16) & 0xFFFFu) | ((tile_x & 0xFFFFu) << 16));
  g1[4] = (int)(tile_y & 0xFFFFu);                        // tile_dim1, tile_dim2=0
  g1[5] = (int)stride_x;                                  // tensor_dim0_stride lo32
  g1[6] = 0;
  g1[7] = 0;
  v4i gz4 = {0, 0, 0, 0};
  v8i gz8 = {0, 0, 0, 0, 0, 0, 0, 0};
  __builtin_amdgcn_tensor_load_to_lds(g0, g1, gz4, gz4, gz8, 0);
#else
  (void)lds_byte; (void)gptr; (void)tile_x; (void)tile_y;
  (void)tensor_x; (void)tensor_y; (void)stride_x; (void)pad_bits;
#endif
}

// pad configs: trigger every 2<<code DWORDs, pad (code+1) DWORDs
static constexpr uint32_t PAD_32_PLUS_2 = (1u << 20) | (4u << 22) | (1u << 25); // 32->34 stride
static constexpr uint32_t PAD_64_PLUS_2 = (1u << 20) | (5u << 22) | (1u << 25); // 64->66 stride

// ---------------------------------------------------------------------------
// Kernel 1: k = E@Wk, v = E@Wv, penalty = relu(cos(E_row, harm) - tau)
// one block (320 threads) per (b, token)
// ---------------------------------------------------------------------------
__global__ void kv_pen_kernel(const float* __restrict__ E, const float* __restrict__ harm,
                              const float* __restrict__ Wk, const float* __restrict__ Wv,
                              float* __restrict__ kp, float* __restrict__ vp,
                              float* __restrict__ pen) {
  const int bt  = blockIdx.x;            // b*KTOK + t
  const int tid = threadIdx.x;           // 0..319 (== output channel)
  __shared__ float Es[DDIM];
  const float* Erow = E + (size_t)bt * DDIM;
  for (int d = tid; d < DDIM; d += blockDim.x) Es[d] = Erow[d];
  __syncthreads();

  float ak = 0.f, av = 0.f;
  for (int d = 0; d < DDIM; ++d) {
    float e = Es[d];
    ak = fmaf(e, Wk[(size_t)d * CH + tid], ak);
    av = fmaf(e, Wv[(size_t)d * CH + tid], av);
  }
  kp[(size_t)bt * CH + tid] = ak;
  vp[(size_t)bt * CH + tid] = av;

  if (tid < 32) {  // first wave: cosine penalty
    float ee = 0.f, eh = 0.f, hh = 0.f;
    for (int d = tid; d < DDIM; d += 32) {
      float e = Es[d], hv = harm[d];
      ee = fmaf(e, e, ee);
      eh = fmaf(e, hv, eh);
      hh = fmaf(hv, hv, hh);
    }
#pragma unroll
    for (int off = 16; off > 0; off >>= 1) {
      ee += __shfl_xor(ee, off, 32);
      eh += __shfl_xor(eh, off, 32);
      hh += __shfl_xor(hh, off, 32);
    }
    if (tid == 0) {
      float den = fmaxf(sqrtf(ee), 1e-12f) * fmaxf(sqrtf(hh), 1e-12f);
      pen[bt] = fmaxf(eh / den - TAU, 0.f);
    }
  }
}

// ---------------------------------------------------------------------------
// Kernel 2: C[M,N] = A[M,K] @ W[K,N] (+bias), fp32 WMMA 16x16x4.
// block = 256 thr (8 waves), tile 64x64, K-chunk 32,
// TDM double-buffered global->LDS staging (TENSORcnt), fallback = coop loads.
// LDS strides: A rows 34 (b64-aligned fragment pairs), B rows 66; both
// conflict-free mod 64 banks for the fragment access patterns.
// ---------------------------------------------------------------------------
__global__ __launch_bounds__(256) void gemm_wmma_kernel(
    const float* __restrict__ A, const float* __restrict__ W,
    const float* __restrict__ bias, float* __restrict__ Out,
    int M, int N, int Kd) {
  __shared__ float Ab[2][64 * 34];
  __shared__ float Bb[2][32 * 66];
  const int mBase = blockIdx.x * 64;
  const int nBase = blockIdx.y * 64;
  const int tid  = threadIdx.x;
  const int w    = tid >> 5;
  const int lane = tid & 31;
  const int lm   = lane & 15;
  const int hi   = lane >> 4;
  const int mi   = w & 3;          // m-tile (0..3)
  const int nj   = (w >> 2) << 1;  // first of two n-tiles (0 or 2)
  const int nChunks = Kd >> 5;

#if HAVE_TDM
  if (tid == 0) {  // prologue prefetch: chunk 0 -> buffer 0
    tdm_load_2d((uint32_t)(uintptr_t)&Ab[0][0], A + (size_t)mBase * Kd,
                32, 64, (uint32_t)Kd, (uint32_t)(M - mBase), (uint32_t)Kd,
                PAD_32_PLUS_2);
    tdm_load_2d((uint32_t)(uintptr_t)&Bb[0][0], W + nBase,
                64, 32, (uint32_t)(N - nBase), (uint32_t)Kd, (uint32_t)N,
                PAD_64_PLUS_2);
  }
#endif

  v8f acc0 = {}, acc1 = {};
  for (int c = 0; c < nChunks; ++c) {
    const int k0  = c << 5;
    const int buf = c & 1;
#if HAVE_TDM
    if (tid == 0) {
      if (c + 1 < nChunks) {  // prefetch next chunk into other buffer
        const int k1 = k0 + 32;
        tdm_load_2d((uint32_t)(uintptr_t)&Ab[buf ^ 1][0],
                    A + (size_t)mBase * Kd + k1,
                    32, 64, (uint32_t)(Kd - k1), (uint32_t)(M - mBase),
                    (uint32_t)Kd, PAD_32_PLUS_2);
        tdm_load_2d((uint32_t)(uintptr_t)&Bb[buf ^ 1][0],
                    W + (size_t)k1 * N + nBase,
                    64, 32, (uint32_t)(N - nBase), (uint32_t)(Kd - k1),
                    (uint32_t)N, PAD_64_PLUS_2);
        WAIT_TENSOR(2);   // current buffer complete; prefetch still in flight
      } else {
        WAIT_TENSOR(0);
      }
    }
    __syncthreads();
#else
    for (int i = tid; i < 64 * 32; i += 256) {
      int m = i >> 5, kk = i & 31;
      Ab[buf][m * 34 + kk] = A[(size_t)(mBase + m) * Kd + k0 + kk];
    }
    for (int i = tid; i < 32 * 64; i += 256) {
      int kk = i >> 6, n = i & 63;
      Bb[buf][kk * 66 + n] = W[(size_t)(k0 + kk) * N + nBase + n];
    }
    __syncthreads();
#endif
#pragma unroll
    for (int kk = 0; kk < 32; kk += 4) {
      v2f a = *(const v2f*)&Ab[buf][(mi * 16 + lm) * 34 + kk + 2 * hi];
      v2f b0, b1;
      b0.x = Bb[buf][(kk + 2 * hi) * 66 + nj * 16 + lm];
      b0.y = Bb[buf][(kk + 2 * hi + 1) * 66 + nj * 16 + lm];
      b1.x = Bb[buf][(kk + 2 * hi) * 66 + nj * 16 + 16 + lm];
      b1.y = Bb[buf][(kk + 2 * hi + 1) * 66 + nj * 16 + 16 + lm];
      acc0 = WMMA_F32(a, b0, acc0);
      acc1 = WMMA_F32(a, b1, acc1);
    }
    __syncthreads();  // all waves done with buf before it is overwritten
  }

#pragma unroll
  for (int j = 0; j < 8; ++j) {
    int row = mBase + mi * 16 + j + 8 * hi;
    int c0  = nBase + nj * 16 + lm;
    int c1  = c0 + 16;
    float v0 = acc0[j], v1 = acc1[j];
    if (bias) { v0 += bias[c0]; v1 += bias[c1]; }
    Out[(size_t)row * N + c0] = v0;
    Out[(size_t)row * N + c1] = v1;
  }
}

// ---------------------------------------------------------------------------
// Kernel 3: attention core. block = 128 thr (4 waves) = 64 queries of one (b,h).
// scores = q@k^T*scale - gamma*pen, softmax over 77 (padded 80), out = P@V
// ---------------------------------------------------------------------------
__global__ __launch_bounds__(128) void attn_kernel(
    const float* __restrict__ q, const float* __restrict__ kp,
    const float* __restrict__ vp, const float* __restrict__ pen,
    float* __restrict__ attn) {
  __shared__ float kT[DH][81];      // K^T: [d][token], padded to 80 tokens
  __shared__ float Vs[80][49];      // V:   [token][d], padded 80x48
  __shared__ float penS[80];
  __shared__ float sc[4][16][82];   // per-wave score/prob tiles (even stride)

  const int bid  = blockIdx.x;
  const int mblk = bid & 63;              // 64 query blocks of 64
  const int h    = (bid >> 6) & 7;
  const int b    = bid >> 9;
  const int tid  = threadIdx.x;
  const int w    = tid >> 5;
  const int lane = tid & 31;
  const int lm   = lane & 15;
  const int hi   = lane >> 4;
  const float scale = 0.15811388300841897f;   // 1/sqrt(40)

  for (int i = tid; i < DH * 80; i += 128) {
    int d = i / 80, t = i % 80;
    kT[d][t] = (t < KTOK) ? kp[((size_t)b * KTOK + t) * CH + h * DH + d] : 0.f;
  }
  for (int i = tid; i < 80 * 48; i += 128) {
    int t = i / 48, d = i % 48;
    Vs[t][d] = (t < KTOK && d < DH) ? vp[((size_t)b * KTOK + t) * CH + h * DH + d] : 0.f;
  }
  for (int i = tid; i < 80; i += 128)
    penS[i] = (i < KTOK) ? GAMMA * pen[b * KTOK + i] : 0.f;
  __syncthreads();

  const int qrow0 = mblk * 64 + w * 16;
  const float* qbase = q + ((size_t)b * SEQ + qrow0) * CH + h * DH;

  // scores: 16x80 = q[16x40] @ kT[40x80]
  v8f s[5] = {};
  for (int kk = 0; kk < DH; kk += 4) {
    v2f a = *(const v2f*)(qbase + (size_t)lm * CH + kk + 2 * hi);  // 8B-aligned
#pragma unroll
    for (int nt = 0; nt < 5; ++nt) {
      v2f bb;
      bb.x = kT[kk + 2 * hi][nt * 16 + lm];
      bb.y = kT[kk + 2 * hi + 1][nt * 16 + lm];
      s[nt] = WMMA_F32(a, bb, s[nt]);
    }
  }
  // scale + penalty + pad-mask -> LDS
#pragma unroll
  for (int nt = 0; nt < 5; ++nt) {
    int col = nt * 16 + lm;
#pragma unroll
    for (int j = 0; j < 8; ++j) {
      float v = s[nt][j] * scale - penS[col];
      if (col >= KTOK) v = NEG_BIG;
      sc[w][j + 8 * hi][col] = v;
    }
  }
  __syncthreads();

  // softmax: one row per lane (lanes 0-15)
  if (lane < 16) {
    float mx = NEG_BIG;
    for (int c = 0; c < 80; ++c) mx = fmaxf(mx, sc[w][lm][c]);
    float sum = 0.f;
    for (int c = 0; c < 80; ++c) {
      float e = __expf(sc[w][lm][c] - mx);
      sc[w][lm][c] = e;
      sum += e;
    }
    float inv = 1.f / sum;
    for (int c = 0; c < 80; ++c) sc[w][lm][c] *= inv;
  }
  __syncthreads();

  // out: 16x48 = P[16x80] @ V[80x48]
  v8f o[3] = {};
  for (int kk = 0; kk < 80; kk += 4) {
    v2f a = *(const v2f*)&sc[w][lm][kk + 2 * hi];
#pragma unroll
    for (int nt = 0; nt < 3; ++nt) {
      v2f bb;
      bb.x = Vs[kk + 2 * hi][nt * 16 + lm];
      bb.y = Vs[kk + 2 * hi + 1][nt * 16 + lm];
      o[nt] = WMMA_F32(a, bb, o[nt]);
    }
  }
#pragma unroll
  for (int nt = 0; nt < 3; ++nt) {
    int col = nt * 16 + lm;
    if (col < DH) {
#pragma unroll
      for (int j = 0; j < 8; ++j) {
        int row = j + 8 * hi;
        attn[((size_t)b * SEQ + qrow0 + row) * CH + h * DH + col] = o[nt][j];
      }
    }
  }
}

// ---------------------------------------------------------------------------
extern "C" void kernel_launch(void* const* d_in, const int* in_sizes, int n_in,
                              void* d_out, int out_size, void* d_ws, size_t ws_size,
                              hipStream_t stream) {
  const float* X    = (const float*)d_in[0];  // [16,4096,320]
  const float* E    = (const float*)d_in[1];  // [16,77,768]
  const float* harm = (const float*)d_in[2];  // [768]
  const float* Wq   = (const float*)d_in[3];  // [320,320]
  const float* Wk   = (const float*)d_in[4];  // [768,320]
  const float* Wv   = (const float*)d_in[5];  // [768,320]
  const float* Wo   = (const float*)d_in[6];  // [320,320]
  const float* bo   = (const float*)d_in[7];  // [320]
  float* out = (float*)d_out;                 // [16,4096,320]; also q scratch
  float* ws  = (float*)d_ws;

  float* kp   = ws;                                    // 16*77*320
  float* vp   = kp + (size_t)BATCH * KTOK * CH;        // 16*77*320
  float* pen  = vp + (size_t)BATCH * KTOK * CH;        // 16*77
  float* attn = pen + (((size_t)BATCH * KTOK + 63) & ~(size_t)63); // 16*4096*320

  const int M = BATCH * SEQ;  // 65536

  // 1) K/V projections + harmful-token penalty
  kv_pen_kernel<<<BATCH * KTOK, CH, 0, stream>>>(E, harm, Wk, Wv, kp, vp, pen);

  // 2) Q = X @ Wq  (into d_out as scratch)
  gemm_wmma_kernel<<<dim3(M / 64, CH / 64), 256, 0, stream>>>(
      X, Wq, nullptr, out, M, CH, CH);

  // 3) attention core -> attn workspace
  attn_kernel<<<BATCH * NH * (SEQ / 64), 128, 0, stream>>>(out, kp, vp, pen, attn);

  // 4) final projection: out = attn @ Wo + bo
  gemm_wmma_kernel<<<dim3(M / 64, CH / 64), 256, 0, stream>>>(
      attn, Wo, bo, out, M, CH, CH);
}